// AlphaKernelConv_89258010346015
// MI455X (gfx1250) — compile-verified
//
#include <hip/hip_runtime.h>

// AlphaKernelConv: out = 3x3 box-sum of (rgb * alpha), zero-padded.
// Bandwidth-bound (117 MB @ 23.3 TB/s ~ 5 us floor). CDNA5 path: TDM
// tensor_load_to_lds (TENSORcnt) DMAs whole 2-D row tiles of all 4 channels
// into LDS; the stencil runs out of LDS with VALU FMAs; outputs stream out
// with non-temporal stores.

static constexpr int IMG_W    = 512;
static constexpr int IMG_H    = 512;
static constexpr int ROWS     = 6;            // output rows per block
static constexpr int SLOTS    = ROWS + 2;     // input rows incl. halo
static constexpr int NTHREADS = 512;          // one thread per column

typedef __attribute__((ext_vector_type(4))) unsigned int v4u;
typedef __attribute__((ext_vector_type(8))) int          v8i;
typedef __attribute__((ext_vector_type(4))) int          v4i;

__global__ __launch_bounds__(NTHREADS)
void alpha_conv_tdm_kernel(const float* __restrict__ x, float* __restrict__ out)
{
    __shared__ float tile[4][SLOTS][IMG_W];   // 4 * 8 * 512 * 4B = 64 KB

    const int j  = threadIdx.x;               // column 0..511
    const int b  = blockIdx.y;                // batch
    const int r0 = blockIdx.x * ROWS;         // first output row of this tile

    // Input rows wanted: r0-1 .. r0+ROWS, clamped to [0, H-1].
    int in_r0 = r0 - 1;
    int slot0 = 0;
    if (in_r0 < 0) { in_r0 = 0; slot0 = 1; }
    int in_r1 = r0 + ROWS;
    if (in_r1 > IMG_H - 1) in_r1 = IMG_H - 1;
    const int nrows = in_r1 - in_r0 + 1;

    // Zero-fill only the halo slots that exist: slot 0 for the top block,
    // trailing slots for bottom/remainder blocks. Interior blocks skip both.
    if (slot0 == 1) {
        #pragma unroll
        for (int c = 0; c < 4; ++c) tile[c][0][j] = 0.0f;
    }
    for (int s = slot0 + nrows; s < SLOTS; ++s) {
        #pragma unroll
        for (int c = 0; c < 4; ++c) tile[c][s][j] = 0.0f;
    }

    // Wave 0 issues one 2-D TDM load per channel (wave-level ops), waits on
    // TENSORcnt, then the workgroup barrier publishes LDS to all 16 waves.
    if (threadIdx.x < 32) {
        // Generic LDS pointer low 32 bits == LDS byte offset (aperture form).
        const unsigned lds_base = (unsigned)(unsigned long long)(&tile[0][0][0]);
        const unsigned d1stride = (unsigned)(IMG_W * IMG_H);  // elements

        for (int c = 0; c < 4; ++c) {
            const unsigned long long ga =
                (unsigned long long)x +
                ((unsigned long long)((b * 4 + c) * IMG_H + in_r0) * IMG_W) * 4ull;
            const unsigned lds_addr =
                lds_base + ((unsigned)(c * SLOTS + slot0) * IMG_W) * 4u;

            // D# group 0 (128b): count=1 | lds_addr | global_addr[56:0] | type=2
            v4u g0;
            g0.x = 1u;                                          // count=1, user mode
            g0.y = lds_addr;                                    // bits[63:32]
            g0.z = (unsigned)(ga & 0xffffffffull);              // global_addr[31:0]
            g0.w = (unsigned)((ga >> 32) & 0x01ffffffull)       // global_addr[56:32]
                 | (2u << 30);                                  // type=2 ("image")

            // D# group 1 (256b): data_size=4B, tensor 512x512, tile 512 x nrows,
            // dim0 stride = 512 elements (contiguous rows).
            v8i g1;
            g1[0] = (int)(2u << 16);                            // data_size=2 (4B), mask=0
            g1[1] = (int)((unsigned)IMG_W << 16);               // tensor_dim0[15:0]=512
            g1[2] = (int)((unsigned)IMG_H << 16);               // tensor_dim1[15:0]=512
            g1[3] = (int)((unsigned)IMG_W << 16);               // tile_dim0=512
            g1[4] = nrows;                                      // tile_dim1=nrows, tile_dim2=0
            g1[5] = IMG_W;                                      // tensor_dim0_stride lo32
            g1[6] = (int)((d1stride & 0xffffu) << 16);          // dim1_stride lo16 (unused, 2D)
            g1[7] = (int)(d1stride >> 16);                      // dim1_stride[47:16]
            v4i g2 = {0, 0, 0, 0};
            v4i g3 = {0, 0, 0, 0};
            v8i g4 = {0, 0, 0, 0, 0, 0, 0, 0};                  // extra state (clang-23 form)

            __builtin_amdgcn_tensor_load_to_lds(g0, g1, g2, g3, g4, 0);
        }
        __builtin_amdgcn_s_wait_tensorcnt(0);
    }
    __syncthreads();

    // Stencil: per input row slot k, horizontal alpha-weighted sums per channel,
    // 3-deep ring per channel for the vertical box sum.
    const int rows_out = (IMG_H - r0 < ROWS) ? (IMG_H - r0) : ROWS;
    float* const outb = out + (unsigned long long)b * 3ull * IMG_H * IMG_W;

    float h0[3], h1[3];

    auto body = [&](int k) {
        const float am = (j > 0)         ? tile[3][k][j - 1] : 0.0f;
        const float a0 =                   tile[3][k][j];
        const float ap = (j < IMG_W - 1) ? tile[3][k][j + 1] : 0.0f;

        float hc[3];
        #pragma unroll
        for (int c = 0; c < 3; ++c) {
            const float xm = (j > 0)         ? tile[c][k][j - 1] : 0.0f;
            const float x0 =                   tile[c][k][j];
            const float xp = (j < IMG_W - 1) ? tile[c][k][j + 1] : 0.0f;
            hc[c] = __builtin_fmaf(xm, am, __builtin_fmaf(x0, a0, xp * ap));
        }
        if (k >= 2) {
            const int row = r0 + k - 2;
            #pragma unroll
            for (int c = 0; c < 3; ++c) {
                __builtin_nontemporal_store(
                    h0[c] + h1[c] + hc[c],
                    &outb[((unsigned long long)c * IMG_H + row) * IMG_W + j]);
            }
        }
        #pragma unroll
        for (int c = 0; c < 3; ++c) { h0[c] = h1[c]; h1[c] = hc[c]; }
    };

    if (rows_out == ROWS) {
        // Common case (85 of 86 row-tiles): fully unrolled 8-slot pipeline.
        #pragma unroll
        for (int k = 0; k < SLOTS; ++k) body(k);
    } else {
        // Bottom remainder tile.
        const int kmax = rows_out + 2;
        for (int k = 0; k < kmax; ++k) body(k);
    }
}

extern "C" void kernel_launch(void* const* d_in, const int* in_sizes, int n_in,
                              void* d_out, int out_size, void* d_ws, size_t ws_size,
                              hipStream_t stream)
{
    const float* x  = (const float*)d_in[0];
    float* outp     = (float*)d_out;

    const int nb = in_sizes[0] / (4 * IMG_H * IMG_W);   // batch = 16
    dim3 grid((IMG_H + ROWS - 1) / ROWS, nb);           // (86, 16) = 1376 blocks
    dim3 block(NTHREADS);
    alpha_conv_tdm_kernel<<<grid, block, 0, stream>>>(x, outp);
}